// HDRClusteringLayer_83588653515400
// MI455X (gfx1250) — compile-verified
//
#include <hip/hip_runtime.h>
#include <stdint.h>

// CDNA5 / gfx1250: wave32, WMMA f32 16x16x4, async global->LDS staging.
typedef float v2f __attribute__((ext_vector_type(2)));
typedef float v4f __attribute__((ext_vector_type(4)));
typedef float v8f __attribute__((ext_vector_type(8)));

#define D_DIM          2048
#define NCLUST         32
#define CHUNK_K        256       // clusters K-chunk staged in LDS (32 KB)
#define BLOCK_THREADS  256       // 8 wave32s
#define ROWS_PER_BLOCK 128       // 8 waves * 16 rows

__global__ __launch_bounds__(BLOCK_THREADS)
void hdr_cluster_q_kernel(const float* __restrict__ inputs,
                          const float* __restrict__ clusters,
                          float* __restrict__ out)
{
    __shared__ float s_cl[NCLUST][CHUNK_K];   // clusters chunk, row-major [n][k]
    __shared__ float s_csq[NCLUST];           // ||c||^2 per cluster

    const int tid  = threadIdx.x;
    const int lane = tid & 31;
    const int wave = tid >> 5;
    const int rowbase = blockIdx.x * ROWS_PER_BLOCK + wave * 16;

    const int m     = lane & 15;         // row within tile (A) / column N (B,C,D)
    const int khalf = (lane >> 4) << 1;  // lanes 0-15 -> K{0,1}; lanes 16-31 -> K{2,3}
    const int hi8   = (lane >> 4) << 3;  // C/D: lanes 16-31 hold M = r + 8

    // ---- cluster squared norms (clusters are tiny -> L2 resident) ----
    if (tid < NCLUST) s_csq[tid] = 0.0f;
    __syncthreads();
    {
        const int n    = tid >> 3;                  // 8 threads per cluster
        const int kseg = (tid & 7) * (D_DIM / 8);
        const float* crow = clusters + n * D_DIM + kseg;
        float p = 0.0f;
        #pragma unroll 4
        for (int i = 0; i < D_DIM / 8; i += 4) {
            v4f c = *(const v4f*)(crow + i);
            p += c.x * c.x + c.y * c.y + c.z * c.z + c.w * c.w;
        }
        atomicAdd(&s_csq[n], p);   // ds_add_f32
    }

    // raw 32-bit LDS offset of s_cl (generic addr low 32 bits == LDS offset)
    const uint32_t lds_base = (uint32_t)(uintptr_t)(&s_cl[0][0]);

    // ---- main GEMM loop: acc = inputs_tile * clusters^T ----
    v8f acc0 = {};                       // clusters 0..15
    v8f acc1 = {};                       // clusters 16..31
    float xacc = 0.0f;                   // per-lane partial of ||x||^2

    const float* arow = inputs + (size_t)(rowbase + m) * D_DIM + khalf;

    for (int kb = 0; kb < D_DIM; kb += CHUNK_K) {
        __syncthreads();
        // async copy clusters[0..31][kb .. kb+CHUNK_K) -> LDS (no VGPR staging)
        #pragma unroll
        for (int j = 0; j < 8; ++j) {
            int f4 = tid + j * BLOCK_THREADS;       // float4 index, 2048 total
            int n  = f4 >> 6;                       // 64 float4s per cluster row
            int k  = (f4 & 63) << 2;
            uint32_t lds   = lds_base + (uint32_t)((n * CHUNK_K + k) * 4);
            uint64_t gaddr = (uint64_t)(uintptr_t)(clusters + (size_t)n * D_DIM + kb + k);
            asm volatile("global_load_async_to_lds_b128 %0, %1, off"
                         :: "v"(lds), "v"(gaddr) : "memory");
        }
        // drain per-wave ASYNCcnt before signaling the workgroup barrier
        asm volatile("s_wait_asynccnt 0x0" ::: "memory");
        __syncthreads();

        if (kb + CHUNK_K < D_DIM)
            __builtin_prefetch(arow + kb + CHUNK_K, 0, 0);   // global_prefetch_b8

        #pragma unroll 4
        for (int s = 0; s < CHUNK_K; s += 4) {
            // A fragment (16x4 f32): lane holds A[m][khalf], A[m][khalf+1]
            v2f a  = *(const v2f*)(arow + kb + s);
            // B fragment (4x16 f32): lane holds B[khalf][m], B[khalf+1][m]
            //   = clusters[n][k]  (k contiguous -> single b64 LDS read)
            v2f b0 = *(const v2f*)&s_cl[m][s + khalf];
            v2f b1 = *(const v2f*)&s_cl[m + 16][s + khalf];

            xacc += a.x * a.x + a.y * a.y;

            acc0 = __builtin_amdgcn_wmma_f32_16x16x4_f32(
                false, a, false, b0, (short)0, acc0, false, false);
            acc1 = __builtin_amdgcn_wmma_f32_16x16x4_f32(
                false, a, false, b1, (short)0, acc1, false, false);
        }
    }

    // fold lane-halves: lane l and l+16 both carry partials for row (l&15)
    float xsq = xacc + __shfl_xor(xacc, 16);

    const float cs0 = s_csq[m];
    const float cs1 = s_csq[m + 16];

    // ---- epilogue: d^2 -> q -> row-normalize -> store ----
    #pragma unroll
    for (int r = 0; r < 8; ++r) {
        // C/D layout: VGPR r, lanes 0-15 -> M=r, N=lane; lanes 16-31 -> M=r+8
        float xs = __shfl(xsq, r + hi8);                 // ||x||^2 of this row
        float d0 = fmaxf(xs - 2.0f * acc0[r] + cs0, 0.0f);
        float d1 = fmaxf(xs - 2.0f * acc1[r] + cs1, 0.0f);
        float q0 = 1.0f / (1.0f + d0);                   // ALPHA = 1, power = 1
        float q1 = 1.0f / (1.0f + d1);

        // row sum over 32 clusters: reduce across 16 lanes of this half
        float ssum = q0 + q1;
        ssum += __shfl_xor(ssum, 1);
        ssum += __shfl_xor(ssum, 2);
        ssum += __shfl_xor(ssum, 4);
        ssum += __shfl_xor(ssum, 8);
        float inv = 1.0f / ssum;

        const int row = rowbase + r + hi8;
        out[row * NCLUST + m]      = q0 * inv;
        out[row * NCLUST + m + 16] = q1 * inv;
    }
}

extern "C" void kernel_launch(void* const* d_in, const int* in_sizes, int n_in,
                              void* d_out, int out_size, void* d_ws, size_t ws_size,
                              hipStream_t stream) {
    (void)n_in; (void)out_size; (void)d_ws; (void)ws_size;
    const float* inputs   = (const float*)d_in[0];
    const float* clusters = (const float*)d_in[1];
    float* out            = (float*)d_out;

    const int nrows  = in_sizes[0] / D_DIM;             // 65536
    const int blocks = nrows / ROWS_PER_BLOCK;          // 512

    hdr_cluster_q_kernel<<<blocks, BLOCK_THREADS, 0, stream>>>(inputs, clusters, out);
}